// GATNodeRegressor_11888469475717
// MI455X (gfx1250) — compile-verified
//
#include <hip/hip_runtime.h>
#include <hip/hip_bf16.h>
#include <math.h>

#define HEADS 4
#define HEAD_DIM 16
#define HID 64
#define NEG_SLOPE 0.2f

typedef __attribute__((ext_vector_type(2))) float v2f;
typedef __attribute__((ext_vector_type(8))) float v8f;

// ---------------------------------------------------------------------------
// C[n,64] = A[n,K] @ B[K,64] via V_WMMA_F32_16X16X4_F32 (fp32 matrix core).
// One wave owns a 16x16 output tile. A-layout (16x4 f32): lanes 0-15 = rows,
// K=0,1 in v0,v1; lanes 16-31 = same rows, K=2,3. B-layout mirrors (4x16).
// C/D: 8 VGPRs, lane l vgpr v -> row v + 8*(l>=16), col l&15.
// ---------------------------------------------------------------------------
__global__ __launch_bounds__(256) void gat_gemm_wmma(
    const float* __restrict__ A, const float* __restrict__ B,
    float* __restrict__ C, int n, int K) {
  const int lane    = threadIdx.x & 31;
  const int wave    = threadIdx.x >> 5;   // 0..7
  const int colTile = wave & 3;           // which 16-col slab of HID=64
  const int rowSub  = wave >> 2;          // 0..1
  const int l15     = lane & 15;
  const int half    = lane >> 4;          // 0 or 1
  const long rowBase = (long)blockIdx.x * 32 + (long)rowSub * 16;

  long arow = rowBase + l15;
  if (arow >= n) arow = n - 1;            // clamp loads; stores are masked
  const float* __restrict__ ap = A + (size_t)arow * K + half * 2;
  const int bcol = colTile * 16 + l15;
  const float* __restrict__ bp = B + (size_t)(half * 2) * HID + bcol;

  v8f acc = {0.f, 0.f, 0.f, 0.f, 0.f, 0.f, 0.f, 0.f};
  for (int k = 0; k < K; k += 4) {
    v2f a; a.x = ap[k];                  a.y = ap[k + 1];
    v2f b; b.x = bp[(size_t)k * HID];    b.y = bp[(size_t)(k + 1) * HID];
    acc = __builtin_amdgcn_wmma_f32_16x16x4_f32(false, a, false, b,
                                                (short)0, acc, false, false);
  }
#pragma unroll
  for (int v = 0; v < 8; ++v) {
    long r = rowBase + v + half * 8;
    if (r < n) C[r * HID + colTile * 16 + l15] = acc[v];
  }
}

// al_s[n,h] = <xh[n,h,:], a_src[h,:]>, al_d likewise. One thread per (node,head).
__global__ void att_logits(const float* __restrict__ xh,
                           const float* __restrict__ asrc,
                           const float* __restrict__ adst,
                           float* __restrict__ als, float* __restrict__ ald,
                           int n) {
  int i = blockIdx.x * blockDim.x + threadIdx.x;
  if (i >= n * HEADS) return;
  int node = i >> 2, h = i & 3;
  const float* xp = xh + (size_t)node * HID + h * HEAD_DIM;
  float s = 0.f, d = 0.f;
#pragma unroll
  for (int c = 0; c < HEAD_DIM; ++c) {
    float v = xp[c];
    s += v * asrc[h * HEAD_DIM + c];
    d += v * adst[h * HEAD_DIM + c];
  }
  als[i] = s;
  ald[i] = d;
}

__global__ void zero_f32(float* __restrict__ p, long cnt) {
  long i = (long)blockIdx.x * blockDim.x + threadIdx.x;
  if (i < cnt) p[i] = 0.f;
}

__device__ __forceinline__ float leaky(float e) {
  return e > 0.f ? e : NEG_SLOPE * e;
}
// Monotonic order-preserving encoding of float for unsigned atomicMax.
__device__ __forceinline__ unsigned enc_f32(float f) {
  unsigned u = __float_as_uint(f);
  return (u & 0x80000000u) ? ~u : (u | 0x80000000u);
}

// Pass 1: segment-max of edge logits into encoded m[dst,h].
__global__ void edge_max(const int* __restrict__ src, const int* __restrict__ dst,
                         const float* __restrict__ als, const float* __restrict__ ald,
                         unsigned* __restrict__ menc, long E, int n) {
  long i = (long)blockIdx.x * blockDim.x + threadIdx.x;
  if (i >= E + n) return;
  int s, d;
  if (i < E) { s = src[i]; d = dst[i]; } else { s = d = (int)(i - E); }
#pragma unroll
  for (int h = 0; h < HEADS; ++h) {
    float e = leaky(als[s * HEADS + h] + ald[d * HEADS + h]);
    atomicMax(&menc[d * HEADS + h], enc_f32(e));
  }
}

// Decode encoded max back to float bits, in place.
__global__ void decode_max(unsigned* __restrict__ m, long cnt) {
  long i = (long)blockIdx.x * blockDim.x + threadIdx.x;
  if (i >= cnt) return;
  unsigned k = m[i];
  m[i] = (k & 0x80000000u) ? (k & 0x7FFFFFFFu) : ~k;
}

// Pass 2: z[dst,h] += exp(e - m[dst,h]).
__global__ void edge_sum(const int* __restrict__ src, const int* __restrict__ dst,
                         const float* __restrict__ als, const float* __restrict__ ald,
                         const float* __restrict__ mf, float* __restrict__ z,
                         long E, int n) {
  long i = (long)blockIdx.x * blockDim.x + threadIdx.x;
  if (i >= E + n) return;
  int s, d;
  if (i < E) { s = src[i]; d = dst[i]; } else { s = d = (int)(i - E); }
#pragma unroll
  for (int h = 0; h < HEADS; ++h) {
    float e = leaky(als[s * HEADS + h] + ald[d * HEADS + h]);
    float p = __expf(e - mf[d * HEADS + h]);
    atomicAdd(&z[d * HEADS + h], p);
  }
}

// Pass 3: agg[dst,h,:] += alpha * xh[src,h,:]. One thread per (edge,head).
__global__ void edge_agg(const int* __restrict__ src, const int* __restrict__ dst,
                         const float* __restrict__ als, const float* __restrict__ ald,
                         const float* __restrict__ mf, const float* __restrict__ z,
                         const float* __restrict__ xh, float* __restrict__ agg,
                         long E, int n) {
  long t = (long)blockIdx.x * blockDim.x + threadIdx.x;
  if (t >= (E + n) * HEADS) return;
  long i = t >> 2;
  int h = (int)(t & 3);
  int s, d;
  if (i < E) { s = src[i]; d = dst[i]; } else { s = d = (int)(i - E); }
  float e = leaky(als[s * HEADS + h] + ald[d * HEADS + h]);
  float alpha = __expf(e - mf[d * HEADS + h]) / (z[d * HEADS + h] + 1e-16f);
  const float* xp = xh + (size_t)s * HID + h * HEAD_DIM;
  float* op = agg + (size_t)d * HID + h * HEAD_DIM;
#pragma unroll
  for (int c = 0; c < HEAD_DIM; ++c) atomicAdd(&op[c], xp[c] * alpha);
}

// h_next = relu(agg + b)
__global__ void finalize_relu(const float* __restrict__ agg,
                              const float* __restrict__ b,
                              float* __restrict__ hout, int n) {
  long i = (long)blockIdx.x * blockDim.x + threadIdx.x;
  if (i >= (long)n * HID) return;
  float v = agg[i] + b[(int)(i & (HID - 1))];
  hout[i] = v > 0.f ? v : 0.f;
}

// y[n] = h[n,:] @ out_w + out_b
__global__ void out_proj(const float* __restrict__ h, const float* __restrict__ ow,
                         const float* __restrict__ ob, float* __restrict__ y, int n) {
  int i = blockIdx.x * blockDim.x + threadIdx.x;
  if (i >= n) return;
  const float* hp = h + (size_t)i * HID;
  float acc = 0.f;
#pragma unroll
  for (int j = 0; j < HID; ++j) acc += hp[j] * ow[j];
  y[i] = acc + ob[0];
}

extern "C" void kernel_launch(void* const* d_in, const int* in_sizes, int n_in,
                              void* d_out, int out_size, void* d_ws, size_t ws_size,
                              hipStream_t stream) {
  const float* x    = (const float*)d_in[0];
  const int*   ei   = (const int*)d_in[1];
  const float* W[3]  = {(const float*)d_in[2], (const float*)d_in[6],  (const float*)d_in[10]};
  const float* As[3] = {(const float*)d_in[3], (const float*)d_in[7],  (const float*)d_in[11]};
  const float* Ad[3] = {(const float*)d_in[4], (const float*)d_in[8],  (const float*)d_in[12]};
  const float* Bs[3] = {(const float*)d_in[5], (const float*)d_in[9],  (const float*)d_in[13]};
  const float* out_w = (const float*)d_in[14];
  const float* out_b = (const float*)d_in[15];
  float* y = (float*)d_out;

  const int  N = in_sizes[0] / 128;
  const long E = in_sizes[1] / 2;
  const int* src = ei;
  const int* dst = ei + E;

  // Workspace layout (floats). agg, mbuf, z are contiguous so a single zero
  // pass of N*72 floats resets all per-layer accumulators.
  float* ws   = (float*)d_ws;
  float* xh   = ws;                        // N*64
  float* agg  = xh   + (size_t)N * HID;    // N*64 (zeroed)
  float* mbuf = agg  + (size_t)N * HID;    // N*4  (zeroed)
  float* z    = mbuf + (size_t)N * HEADS;  // N*4  (zeroed)
  float* hbuf = z    + (size_t)N * HEADS;  // N*64
  float* als  = hbuf + (size_t)N * HID;    // N*4
  float* ald  = als  + (size_t)N * HEADS;  // N*4

  const long ES  = E + N;                  // edges incl. self-loops
  const int  nh  = N * HEADS;
  const long zc  = (long)N * (HID + 2 * HEADS);

  for (int l = 0; l < 3; ++l) {
    const float* hin = (l == 0) ? x : hbuf;
    const int K = (l == 0) ? 128 : HID;

    gat_gemm_wmma<<<dim3((N + 31) / 32), 256, 0, stream>>>(hin, W[l], xh, N, K);
    att_logits<<<dim3((nh + 255) / 256), 256, 0, stream>>>(xh, As[l], Ad[l], als, ald, N);
    zero_f32<<<dim3((unsigned)((zc + 255) / 256)), 256, 0, stream>>>(agg, zc);
    edge_max<<<dim3((unsigned)((ES + 255) / 256)), 256, 0, stream>>>(
        src, dst, als, ald, (unsigned*)mbuf, E, N);
    decode_max<<<dim3((nh + 255) / 256), 256, 0, stream>>>((unsigned*)mbuf, nh);
    edge_sum<<<dim3((unsigned)((ES + 255) / 256)), 256, 0, stream>>>(
        src, dst, als, ald, mbuf, z, E, N);
    edge_agg<<<dim3((unsigned)((ES * HEADS + 255) / 256)), 256, 0, stream>>>(
        src, dst, als, ald, mbuf, z, xh, agg, E, N);
    finalize_relu<<<dim3((unsigned)(((long)N * HID + 255) / 256)), 256, 0, stream>>>(
        agg, Bs[l], hbuf, N);
  }

  out_proj<<<dim3((N + 255) / 256), 256, 0, stream>>>(hbuf, out_w, out_b, y, N);
}